// HeteroConvLayer2_56581899157982
// MI455X (gfx1250) — compile-verified
//
#include <hip/hip_runtime.h>

#define DD 128

typedef __attribute__((ext_vector_type(2))) float v2f;
typedef __attribute__((ext_vector_type(8))) float v8f;
typedef __attribute__((ext_vector_type(4))) unsigned int u32x4;
typedef __attribute__((ext_vector_type(8))) unsigned int u32x8;

// ---------------- prep: compose the chained linears ----------------
// T1w = W_wt x W_ww ;  Mt = (W_tt x W_td)^T   (so ht = h_topic @ Mt + ct)
__global__ void prep_mm1(const float* __restrict__ Wwt, const float* __restrict__ Www,
                         const float* __restrict__ Wtt, const float* __restrict__ Wtd,
                         float* __restrict__ T1w, float* __restrict__ Mt) {
    const float* A = (blockIdx.x == 0) ? Wwt : Wtt;
    const float* B = (blockIdx.x == 0) ? Www : Wtd;
    for (int idx = threadIdx.x; idx < DD * DD; idx += blockDim.x) {
        int i = idx / DD, j = idx % DD;
        float s = 0.f;
        for (int k = 0; k < DD; ++k) s = fmaf(A[i * DD + k], B[k * DD + j], s);
        if (blockIdx.x == 0) T1w[i * DD + j] = s;
        else                 Mt[j * DD + i] = s;   // store transposed
    }
}

// Mw = (W_wd x T1w)^T ; cw = W_wd*(W_wt*b_ww + b_wt) + b_wd ; ct = W_tt*b_td + b_tt
__global__ void prep_mm2(const float* __restrict__ Wwd, const float* __restrict__ T1w,
                         float* __restrict__ Mw,
                         const float* __restrict__ Wwt, const float* __restrict__ b_ww,
                         const float* __restrict__ b_wt, const float* __restrict__ b_wd,
                         const float* __restrict__ Wtt, const float* __restrict__ b_td,
                         const float* __restrict__ b_tt,
                         float* __restrict__ cw, float* __restrict__ ct) {
    if (blockIdx.x == 0) {
        for (int idx = threadIdx.x; idx < DD * DD; idx += blockDim.x) {
            int i = idx / DD, j = idx % DD;
            float s = 0.f;
            for (int k = 0; k < DD; ++k) s = fmaf(Wwd[i * DD + k], T1w[k * DD + j], s);
            Mw[j * DD + i] = s;                    // store transposed
        }
    } else {
        __shared__ float bw2[DD];
        int t = threadIdx.x;
        if (t < DD) {
            float s = b_wt[t];
            for (int i = 0; i < DD; ++i) s = fmaf(Wwt[t * DD + i], b_ww[i], s);
            bw2[t] = s;
        }
        __syncthreads();
        if (t < DD) {
            float s = b_wd[t];
            for (int j = 0; j < DD; ++j) s = fmaf(Wwd[t * DD + j], bw2[j], s);
            cw[t] = s;
        } else if (t < 2 * DD) {
            int n = t - DD;
            float s = b_tt[n];
            for (int j = 0; j < DD; ++j) s = fmaf(Wtt[n * DD + j], b_td[j], s);
            ct[n] = s;
        }
    }
}

// ---------------- fp32 WMMA GEMM: Y = X @ Mc + bias ----------------
// 4 waves/block, 16 rows per wave, K=128 via 32 chained 16x16x4 fp32 WMMAs per N-tile.
// Weight tile (128x128 fp32, 64 KB) staged global->LDS by the Tensor Data Mover.
__global__ void __launch_bounds__(128)
gemm_wmma_f32(const float* __restrict__ X, const float* __restrict__ Mc,
              const float* __restrict__ bias, float* __restrict__ Y, int nrows) {
    extern __shared__ float ldsW[];                 // DD*DD fp32 = 64 KB at LDS offset 0

    const int lane = threadIdx.x & 31;
    const int wave = threadIdx.x >> 5;

    if (wave == 0) {
        // D# descriptor (CDNA5 ISA): 2-D tensor, whole
        // 128x128 tile, 4-byte elements, dim0 stride = 128 elements.
        const unsigned long long ga = (unsigned long long)(const void*)Mc;
        u32x4 g0;
        g0[0] = 1u;                                   // count=1 (valid), no gather
        g0[1] = 0u;                                   // lds_addr = 0
        g0[2] = (unsigned int)ga;                     // global_addr[31:0]
        g0[3] = (unsigned int)((ga >> 32) & 0x01FFFFFFu) | (2u << 30); // addr[56:32] | type=2
        u32x8 g1;
        g1[0] = 2u << 16;                             // workgroup_mask=0, data_size=2 (4B)
        g1[1] = (unsigned int)DD << 16;               // tensor_dim0[15:0]=128
        g1[2] = (unsigned int)DD << 16;               // tensor_dim1[15:0]=128
        g1[3] = (unsigned int)DD << 16;               // tile_dim0=128
        g1[4] = (unsigned int)DD;                     // tile_dim1=128, tile_dim2=0
        g1[5] = (unsigned int)DD;                     // tensor_dim0_stride[31:0]=128
        g1[6] = 0u;                                   // stride hi / dim1_stride (unused, 2D)
        g1[7] = 0u;
        asm volatile("tensor_load_to_lds %0, %1" :: "s"(g0), "s"(g1) : "memory");
        __builtin_amdgcn_s_wait_tensorcnt(0);
    }
    __syncthreads();

    const int row_base = (blockIdx.x * 4 + wave) * 16;
    if (row_base >= nrows) return;                  // wave-uniform exit
    const int lrow  = lane & 15;
    const int khalf = (lane >> 4) << 1;             // 0 or 2 (A/B K sub-slice)
    const int mbase = (lane >> 4) << 3;             // 0 or 8 (C/D M half)
    int row = row_base + lrow;
    if (row >= nrows) row = nrows - 1;              // clamp loads, mask stores
    const float* xr = X + (size_t)row * DD;

    v2f a[32];                                      // whole 16x128 A tile in regs
#pragma unroll
    for (int kk = 0; kk < 32; ++kk) {
        const int k0 = kk * 4 + khalf;
        a[kk].x = xr[k0];
        a[kk].y = xr[k0 + 1];
    }
    float bv[8];                                    // hoist bias out of the WMMA stream
#pragma unroll
    for (int nt = 0; nt < 8; ++nt) bv[nt] = bias[nt * 16 + lrow];

    for (int nt = 0; nt < 8; ++nt) {
        v8f c = {};
#pragma unroll
        for (int kk = 0; kk < 32; ++kk) {
            const int k0 = kk * 4 + khalf;
            v2f b;
            b.x = ldsW[k0 * DD + nt * 16 + lrow];
            b.y = ldsW[(k0 + 1) * DD + nt * 16 + lrow];
            c = __builtin_amdgcn_wmma_f32_16x16x4_f32(
                    false, a[kk], false, b, (short)0, c, false, false);
        }
#pragma unroll
        for (int r = 0; r < 8; ++r) {
            const int orow = row_base + mbase + r;
            if (orow < nrows)
                Y[(size_t)orow * DD + nt * 16 + lrow] = c[r] + bv[nt];
        }
    }
}

// ---------------- graph side ----------------
__global__ void zero_f32(float* __restrict__ p, long n) {
    long i = (long)blockIdx.x * blockDim.x + threadIdx.x;
    long stride = (long)gridDim.x * blockDim.x;
    for (; i < n; i += stride) p[i] = 0.f;
}

__global__ void count_edges(const int* __restrict__ dst, float* __restrict__ cnt, int E) {
    int e = blockIdx.x * blockDim.x + threadIdx.x;
    if (e < E) atomicAdd(&cnt[dst[e]], 1.0f);
}

// out[dst] += h[src] * (ew / max(cnt[dst],1)) — 32 lanes per edge, float4 per lane
__global__ void scatter_mean(const float* __restrict__ H, const int* __restrict__ src,
                             const int* __restrict__ dst, const float* __restrict__ ew,
                             const float* __restrict__ cnt, float* __restrict__ out, int E) {
    long gid = (long)blockIdx.x * blockDim.x + threadIdx.x;
    long e = gid >> 5;
    if (e >= E) return;
    int q = (int)(gid & 31);
    int s = src[e], d0 = dst[e];
    float scale = ew[e] / fmaxf(cnt[d0], 1.0f);
    const float4 v = ((const float4*)(H + (size_t)s * DD))[q];
    float* o = out + (size_t)d0 * DD + q * 4;
    atomicAdd(o + 0, v.x * scale);
    atomicAdd(o + 1, v.y * scale);
    atomicAdd(o + 2, v.z * scale);
    atomicAdd(o + 3, v.w * scale);
}

__global__ void relu_f32(float* __restrict__ p, long n) {
    long i = (long)blockIdx.x * blockDim.x + threadIdx.x;
    long stride = (long)gridDim.x * blockDim.x;
    for (; i < n; i += stride) p[i] = fmaxf(p[i], 0.f);
}

extern "C" void kernel_launch(void* const* d_in, const int* in_sizes, int n_in,
                              void* d_out, int out_size, void* d_ws, size_t ws_size,
                              hipStream_t stream) {
    const float* h_word  = (const float*)d_in[0];
    const float* h_topic = (const float*)d_in[1];
    // d_in[2] = h_doc: only its row count is used
    const float *W_ww = (const float*)d_in[3],  *b_ww = (const float*)d_in[4];
    const float *W_wt = (const float*)d_in[5],  *b_wt = (const float*)d_in[6];
    const float *W_wd = (const float*)d_in[7],  *b_wd = (const float*)d_in[8];
    const float *W_td = (const float*)d_in[9],  *b_td = (const float*)d_in[10];
    const float *W_tt = (const float*)d_in[11], *b_tt = (const float*)d_in[12];
    const int *src_ww = (const int*)d_in[13], *dst_ww = (const int*)d_in[14];
    const float* ew_ww = (const float*)d_in[15];
    const int *src_wt = (const int*)d_in[16], *dst_wt = (const int*)d_in[17];
    const float* ew_wt = (const float*)d_in[18];
    const int *src_wd = (const int*)d_in[19], *dst_wd = (const int*)d_in[20];
    const float* ew_wd = (const float*)d_in[21];
    const int *src_td = (const int*)d_in[22], *dst_td = (const int*)d_in[23];
    const float* ew_td = (const float*)d_in[24];
    const int *src_tt = (const int*)d_in[25], *dst_tt = (const int*)d_in[26];
    const float* ew_tt = (const float*)d_in[27];

    const int NW = in_sizes[0] / DD;
    const int NT = in_sizes[1] / DD;
    const int ND = in_sizes[2] / DD;
    const int Eww = in_sizes[13], Ewt = in_sizes[16], Ewd = in_sizes[19];
    const int Etd = in_sizes[22], Ett = in_sizes[25];

    // workspace layout (floats)
    float* ws  = (float*)d_ws;
    float* Mw  = ws;                          // 16384
    float* Mt  = Mw + DD * DD;                // 16384
    float* T1w = Mt + DD * DD;                // 16384
    float* cw  = T1w + DD * DD;               // 128
    float* ct  = cw + DD;                     // 128
    float* hw  = ct + DD;                     // NW*128
    float* ht  = hw + (size_t)NW * DD;        // NT*128
    float* cnt_ww = ht + (size_t)NT * DD;     // NW
    float* cnt_wt = cnt_ww + NW;              // NT
    float* cnt_tt = cnt_wt + NT;              // NT
    float* cnt_wd = cnt_tt + NT;              // ND
    float* cnt_td = cnt_wd + ND;              // ND
    const long cnt_total = (long)NW + 2L * NT + 2L * ND;

    float* out_word  = (float*)d_out;
    float* out_topic = out_word + (size_t)NW * DD;
    float* out_doc   = out_topic + (size_t)NT * DD;

    auto cdiv = [](long a, long b) { return (int)((a + b - 1) / b); };

    // 1) compose weights/biases
    prep_mm1<<<2, 256, 0, stream>>>(W_wt, W_ww, W_tt, W_td, T1w, Mt);
    prep_mm2<<<2, 256, 0, stream>>>(W_wd, T1w, Mw, W_wt, b_ww, b_wt, b_wd,
                                    W_tt, b_td, b_tt, cw, ct);

    // 2) zero accumulators
    zero_f32<<<2048, 256, 0, stream>>>((float*)d_out, (long)out_size);
    zero_f32<<<512, 256, 0, stream>>>(cnt_ww, cnt_total);

    // 3) degree counts per relation
    count_edges<<<cdiv(Eww, 256), 256, 0, stream>>>(dst_ww, cnt_ww, Eww);
    count_edges<<<cdiv(Ewt, 256), 256, 0, stream>>>(dst_wt, cnt_wt, Ewt);
    count_edges<<<cdiv(Ewd, 256), 256, 0, stream>>>(dst_wd, cnt_wd, Ewd);
    count_edges<<<cdiv(Etd, 256), 256, 0, stream>>>(dst_td, cnt_td, Etd);
    count_edges<<<cdiv(Ett, 256), 256, 0, stream>>>(dst_tt, cnt_tt, Ett);

    // 4) WMMA GEMMs: hw = h_word@Mw+cw, ht = h_topic@Mt+ct
    const size_t shmem = (size_t)DD * DD * sizeof(float);   // 64 KB
    gemm_wmma_f32<<<cdiv(NW, 64), 128, shmem, stream>>>(h_word, Mw, cw, hw, NW);
    gemm_wmma_f32<<<cdiv(NT, 64), 128, shmem, stream>>>(h_topic, Mt, ct, ht, NT);

    // 5) mean aggregation (1/cnt folded into edge weight), cross-relation sum
    scatter_mean<<<cdiv(32L * Eww, 256), 256, 0, stream>>>(hw, src_ww, dst_ww, ew_ww, cnt_ww, out_word, Eww);
    scatter_mean<<<cdiv(32L * Ewt, 256), 256, 0, stream>>>(hw, src_wt, dst_wt, ew_wt, cnt_wt, out_topic, Ewt);
    scatter_mean<<<cdiv(32L * Ett, 256), 256, 0, stream>>>(ht, src_tt, dst_tt, ew_tt, cnt_tt, out_topic, Ett);
    scatter_mean<<<cdiv(32L * Ewd, 256), 256, 0, stream>>>(hw, src_wd, dst_wd, ew_wd, cnt_wd, out_doc, Ewd);
    scatter_mean<<<cdiv(32L * Etd, 256), 256, 0, stream>>>(ht, src_td, dst_td, ew_td, cnt_td, out_doc, Etd);

    // 6) ReLU
    relu_f32<<<2048, 256, 0, stream>>>((float*)d_out, (long)out_size);

    (void)n_in; (void)ws_size;
}